// YOLOLayer_29454885716156
// MI455X (gfx1250) — compile-verified
//
#include <hip/hip_runtime.h>
#include <cstdint>

#define G        76
#define GG       (76 * 76)          // 5776
#define NA       3                  // anchors
#define CPA      85                 // 5 + 80 channels per anchor
#define CH_TOT   (NA * CPA)         // 255
#define STRIDE_F 8.0f               // 608 / 76
#define TILE_FLOATS (G * CPA)       // 6460 floats per (b, a, gy) row-chunk
#define TILE_VEC4   (TILE_FLOATS / 4) // 1615 16-byte packets (exact)
#define NTHREADS 256

__device__ __forceinline__ float fast_sigmoid(float x) {
    return 1.0f / (1.0f + __expf(-x));
}

__global__ __launch_bounds__(NTHREADS) void yolo_layer_kernel(
    const float* __restrict__ in, float* __restrict__ out)
{
    // LDS tile holds the exact byte image of this block's output chunk:
    // tile[p * 85 + c]  ==  out[chunk_base + p*85 + c]
    __shared__ __align__(16) float tile[TILE_FLOATS];

    const int m  = blockIdx.x;        // m = (b*NA + a)*G + gy
    const int gy = m % G;
    const int ba = m / G;
    const int a  = ba % NA;
    const int b  = ba / NA;

    // W = exp(v) * (anchor/stride) * stride  ==  exp(v) * anchor
    const float aw  = (a == 0) ? 10.0f : (a == 1) ? 16.0f : 33.0f;
    const float ah  = (a == 0) ? 13.0f : (a == 1) ? 30.0f : 23.0f;
    const float gyf = (float)gy;

    // input base for (b, a, c=0, gy, gx=0); channel c advances by GG floats
    const float* __restrict__ src =
        in + (size_t)(b * CH_TOT + a * CPA) * GG + (size_t)gy * G;
    float* __restrict__ dst = out + (size_t)m * TILE_FLOATS;

    const int tid = threadIdx.x;

    // ---- Phase 1: coalesced b128 loads along spatial axis, transform,
    //               transpose-scatter into LDS (stride 85 -> odd, low conflict)
    for (int q = tid; q < CPA * (G / 4); q += NTHREADS) {   // 85 * 19 = 1615
        const int c  = q / (G / 4);
        const int p4 = (q - c * (G / 4)) * 4;               // 16B-aligned (76 % 4 == 0)
        const float4 v4 = *(const float4*)(src + (size_t)c * GG + p4);
        float vv[4] = {v4.x, v4.y, v4.z, v4.w};
        #pragma unroll
        for (int k = 0; k < 4; ++k) {
            const int p = p4 + k;
            const float v = vv[k];
            float t;
            if (c >= 4) {                 // objectness + 80 class scores
                t = fast_sigmoid(v);
            } else if (c == 0) {          // X = (sigmoid + gx) * stride
                t = (fast_sigmoid(v) + (float)p) * STRIDE_F;
            } else if (c == 1) {          // Y = (sigmoid + gy) * stride
                t = (fast_sigmoid(v) + gyf) * STRIDE_F;
            } else if (c == 2) {          // W = exp * anchor_w
                t = __expf(v) * aw;
            } else {                      // H = exp * anchor_h
                t = __expf(v) * ah;
            }
            tile[p * CPA + c] = t;
        }
    }

    __syncthreads();

    // ---- Phase 2: CDNA5 async DMA LDS -> global, 16B per lane, fully
    //               coalesced & aligned (chunk base = m * 25840 B, 16B multiple)
    const unsigned lbase = (unsigned)(size_t)(&tile[0]);  // wave-relative LDS byte addr
    for (int i = tid; i < TILE_VEC4; i += NTHREADS) {
        const uint64_t gaddr = (uint64_t)(dst + (size_t)i * 4);
        const unsigned laddr = lbase + (unsigned)(i * 16);
        asm volatile("global_store_async_from_lds_b128 %0, %1, off"
                     :: "v"(gaddr), "v"(laddr)
                     : "memory");
    }
    // Drain ASYNCcnt before the workgroup (and its LDS) can retire.
    asm volatile("s_wait_asynccnt 0" ::: "memory");
}

extern "C" void kernel_launch(void* const* d_in, const int* in_sizes, int n_in,
                              void* d_out, int out_size, void* d_ws, size_t ws_size,
                              hipStream_t stream) {
    const float* in = (const float*)d_in[0];
    float* out      = (float*)d_out;

    // Derive batch from input size: B * 255 * 76 * 76 elements.
    const int B = in_sizes[0] / (CH_TOT * GG);
    const int nblocks = B * NA * G;      // one block per (b, anchor, gy) row

    yolo_layer_kernel<<<dim3(nblocks), dim3(NTHREADS), 0, stream>>>(in, out);
}